// GNN_Net_14989435863229
// MI455X (gfx1250) — compile-verified
//
#include <hip/hip_runtime.h>
#include <hip/hip_bf16.h>

typedef __attribute__((ext_vector_type(2))) float v2f;
typedef __attribute__((ext_vector_type(8))) float v8f;

#define N_NODES  50000
#define N_EDGES  1600000
#define N_GRAPHS 512
#define EPS      1e-5f

__device__ __forceinline__ void atomicAddF(float* p, float v) {
#if defined(__AMDGCN__)
    unsafeAtomicAdd(p, v);   // lowers to global_atomic_add_f32 on gfx1250
#else
    atomicAdd(p, v);
#endif
}

// ---------------- zero fill ----------------
__global__ void k_zero(float* __restrict__ p, int n) {
    int t = blockIdx.x * blockDim.x + threadIdx.x;
    for (; t < n; t += gridDim.x * blockDim.x) p[t] = 0.0f;
}

// ---------------- node embedding: h = x @ W_node + b_node  [N,9]x[9,64] ----------------
__global__ void k_node_embed(const float* __restrict__ x, const float* __restrict__ Wn,
                             const float* __restrict__ bn, float* __restrict__ h) {
    int t = blockIdx.x * blockDim.x + threadIdx.x;
    if (t >= N_NODES * 64) return;
    int n = t >> 6, c = t & 63;
    float acc = bn[c];
    const float* xr = x + n * 9;
    #pragma unroll
    for (int k = 0; k < 9; ++k) acc += xr[k] * Wn[k * 64 + c];
    h[t] = acc;
}

// ---------------- edge weights: w = edge_attr @ W_edge + b_edge ----------------
__global__ void k_edge_w(const float* __restrict__ ea, const float* __restrict__ We,
                         const float* __restrict__ be, float* __restrict__ w) {
    int e = blockIdx.x * blockDim.x + threadIdx.x;
    if (e >= N_EDGES) return;
    const float* a = ea + e * 3;
    w[e] = a[0] * We[0] + a[1] * We[1] + a[2] * We[2] + be[0];
}

// ---------------- scatter: agg[dst] += h[src] * w   (d features, float4 per thread) ----------------
__global__ void k_scatter(const float* __restrict__ h, const float* __restrict__ w,
                          const int* __restrict__ src, const int* __restrict__ dst,
                          float* __restrict__ agg, int d, int lg2groups) {
    long long t = (long long)blockIdx.x * blockDim.x + threadIdx.x;
    int groups = 1 << lg2groups;                 // d/4
    long long total = (long long)N_EDGES << lg2groups;
    if (t >= total) return;
    int e = (int)(t >> lg2groups);
    int f = (int)(t & (groups - 1)) << 2;
    int s = src[e], q = dst[e];
    float we = w[e];
    float4 v = *(const float4*)(h + (long long)s * d + f);
    float* base = agg + (long long)q * d + f;
    atomicAddF(base + 0, v.x * we);
    atomicAddF(base + 1, v.y * we);
    atomicAddF(base + 2, v.z * we);
    atomicAddF(base + 3, v.w * we);
}

// ---------------- dual-source GEMM via V_WMMA_F32_16X16X4_F32 ----------------
// out[M,N] = A1[M,K] @ W1[K,N] + A2[M,K] @ W2[K,N] + bias[N]
// one wave computes one 16x16 tile; M must be a multiple of 16 (50000 = 3125*16)
__global__ void k_gemm_dual_wmma(const float* __restrict__ A1, const float* __restrict__ W1,
                                 const float* __restrict__ A2, const float* __restrict__ W2,
                                 const float* __restrict__ bias, float* __restrict__ out,
                                 int M, int K, int N) {
    int wave = (blockIdx.x * blockDim.x + threadIdx.x) >> 5;
    int lane = threadIdx.x & 31;
    int ntiles = N >> 4;
    int tileN = wave % ntiles;
    int tileM = wave / ntiles;
    if (tileM * 16 >= M) return;                 // wave-uniform: EXEC stays all-ones

    int half = lane >> 4;                        // 0: K+{0,1}, 1: K+{2,3}
    int l    = lane & 15;
    int row  = tileM * 16 + l;                   // A row for this lane
    int col  = tileN * 16 + l;                   // B/C column for this lane

    v8f c = {};
    for (int k = 0; k < K; k += 4) {
        int kk = k + half * 2;
        v2f a = *(const v2f*)(A1 + (long long)row * K + kk);
        v2f b;
        b.x = W1[(long long)kk * N + col];
        b.y = W1[(long long)(kk + 1) * N + col];
        c = __builtin_amdgcn_wmma_f32_16x16x4_f32(false, a, false, b, (short)0, c, false, false);
    }
    for (int k = 0; k < K; k += 4) {
        int kk = k + half * 2;
        v2f a = *(const v2f*)(A2 + (long long)row * K + kk);
        v2f b;
        b.x = W2[(long long)kk * N + col];
        b.y = W2[(long long)(kk + 1) * N + col];
        c = __builtin_amdgcn_wmma_f32_16x16x4_f32(false, a, false, b, (short)0, c, false, false);
    }

    float bcol = bias[col];
    #pragma unroll
    for (int r = 0; r < 8; ++r) {
        int orow = tileM * 16 + r + half * 8;    // C/D layout: lanes 16-31 hold rows +8
        out[(long long)orow * N + col] = c[r] + bcol;
    }
}

// ---------------- BatchNorm stats: stats[0:C]=sum, stats[C:2C]=sumsq ----------------
__global__ void k_bn_stats(const float* __restrict__ in, float* __restrict__ stats,
                           int nrows, int ncols) {
    int tid = threadIdx.x;
    int col = tid % ncols;
    int rpb = blockDim.x / ncols;                // rows walked per block per step
    int r   = blockIdx.x * rpb + tid / ncols;
    int stride = gridDim.x * rpb;
    float s = 0.0f, ss = 0.0f;
    for (; r < nrows; r += stride) {
        float v = in[(long long)r * ncols + col];
        s += v; ss += v * v;
    }
    atomicAddF(&stats[col], s);
    atomicAddF(&stats[ncols + col], ss);
}

// ---------------- BatchNorm apply + ReLU (in place) ----------------
__global__ void k_bn_relu(float* __restrict__ io, const float* __restrict__ stats,
                          const float* __restrict__ gamma, const float* __restrict__ beta,
                          int total, int colMask, int ncols, float invN) {
    int t = blockIdx.x * blockDim.x + threadIdx.x;
    if (t >= total) return;
    int c = t & colMask;
    float mean = stats[c] * invN;
    float var  = stats[ncols + c] * invN - mean * mean;
    float scale = gamma[c] * rsqrtf(var + EPS);
    float v = (io[t] - mean) * scale + beta[c];
    io[t] = fmaxf(v, 0.0f);
}

// ---------------- global mean pool: accumulate sums + counts ----------------
__global__ void k_pool(const float* __restrict__ h2, const int* __restrict__ batch,
                       float* __restrict__ pool, float* __restrict__ counts) {
    int t = blockIdx.x * blockDim.x + threadIdx.x;
    if (t >= N_NODES * 32) return;
    int n = t >> 5, c = t & 31;
    int g = batch[n];
    atomicAddF(&pool[g * 32 + c], h2[t]);
    if (c == 0) atomicAddF(&counts[g], 1.0f);
}

// ---------------- MLP head: pred = relu(gx @ W1 + b1) @ W2 + b2 ----------------
__global__ void k_head(const float* __restrict__ pool, const float* __restrict__ counts,
                       const float* __restrict__ W1, const float* __restrict__ b1,
                       const float* __restrict__ W2, const float* __restrict__ b2,
                       float* __restrict__ out) {
    int g = blockIdx.x * blockDim.x + threadIdx.x;
    if (g >= N_GRAPHS) return;
    float inv = 1.0f / fmaxf(counts[g], 1.0f);
    float gx[32];
    #pragma unroll
    for (int c = 0; c < 32; ++c) gx[c] = pool[g * 32 + c] * inv;
    float hid[16];
    #pragma unroll
    for (int j = 0; j < 16; ++j) {
        float a = b1[j];
        #pragma unroll
        for (int c = 0; c < 32; ++c) a += gx[c] * W1[c * 16 + j];
        hid[j] = fmaxf(a, 0.0f);
    }
    #pragma unroll
    for (int o = 0; o < 2; ++o) {
        float a = b2[o];
        #pragma unroll
        for (int j = 0; j < 16; ++j) a += hid[j] * W2[j * 2 + o];
        out[g * 2 + o] = a;
    }
}

extern "C" void kernel_launch(void* const* d_in, const int* in_sizes, int n_in,
                              void* d_out, int out_size, void* d_ws, size_t ws_size,
                              hipStream_t stream) {
    const float* x        = (const float*)d_in[0];
    const float* ea       = (const float*)d_in[1];
    const int*   eidx     = (const int*)  d_in[2];
    const int*   batch    = (const int*)  d_in[3];
    const float* W_node   = (const float*)d_in[4];
    const float* b_node   = (const float*)d_in[5];
    const float* W_edge   = (const float*)d_in[6];
    const float* b_edge   = (const float*)d_in[7];
    const float* Wrel1    = (const float*)d_in[8];
    const float* brel1    = (const float*)d_in[9];
    const float* Wroot1   = (const float*)d_in[10];
    const float* gamma1   = (const float*)d_in[11];
    const float* beta1    = (const float*)d_in[12];
    const float* Wrel2    = (const float*)d_in[13];
    const float* brel2    = (const float*)d_in[14];
    const float* Wroot2   = (const float*)d_in[15];
    const float* gamma2   = (const float*)d_in[16];
    const float* beta2    = (const float*)d_in[17];
    const float* W1       = (const float*)d_in[18];
    const float* b1       = (const float*)d_in[19];
    const float* W2       = (const float*)d_in[20];
    const float* b2       = (const float*)d_in[21];
    float* out = (float*)d_out;

    const int* src = eidx;
    const int* dst = eidx + N_EDGES;

    // workspace layout (floats)
    float* ws    = (float*)d_ws;
    float* h     = ws;                       // 50000*64  = 3,200,000
    float* wE    = h    + 3200000;           // 1,600,000
    float* agg1  = wE   + 1600000;           // 3,200,000
    float* out1  = agg1 + 3200000;           // 50000*128 = 6,400,000
    float* agg2  = out1 + 6400000;           // 6,400,000
    float* out2  = agg2 + 6400000;           // 50000*32  = 1,600,000
    float* st1   = out2 + 1600000;           // 256
    float* st2   = st1  + 256;               // 64
    float* pool  = st2  + 64;                // 512*32 = 16384
    float* cnts  = pool + 16384;             // 512

    const int B = 256;

    // zero accumulators (harness does not re-zero between replays)
    k_zero<<<12500, B, 0, stream>>>(agg1, 3200000);
    k_zero<<<25000, B, 0, stream>>>(agg2, 6400000);
    k_zero<<<68,    B, 0, stream>>>(st1, 256 + 64 + 16384 + 512);

    // embeddings
    k_node_embed<<<(N_NODES * 64 + B - 1) / B, B, 0, stream>>>(x, W_node, b_node, h);
    k_edge_w<<<(N_EDGES + B - 1) / B, B, 0, stream>>>(ea, W_edge, b_edge, wE);

    // GraphConv 1: scatter + dual GEMM (WMMA f32) + BN + ReLU
    k_scatter<<<(int)(((long long)N_EDGES * 16 + B - 1) / B), B, 0, stream>>>(
        h, wE, src, dst, agg1, 64, 4);
    {
        int waves = (N_NODES / 16) * (128 / 16);           // 25000
        k_gemm_dual_wmma<<<(waves * 32 + B - 1) / B, B, 0, stream>>>(
            agg1, Wrel1, h, Wroot1, brel1, out1, N_NODES, 64, 128);
    }
    k_bn_stats<<<128, B, 0, stream>>>(out1, st1, N_NODES, 128);
    k_bn_relu<<<(N_NODES * 128 + B - 1) / B, B, 0, stream>>>(
        out1, st1, gamma1, beta1, N_NODES * 128, 127, 128, 1.0f / N_NODES);

    // GraphConv 2
    k_scatter<<<(int)(((long long)N_EDGES * 32 + B - 1) / B), B, 0, stream>>>(
        out1, wE, src, dst, agg2, 128, 5);
    {
        int waves = (N_NODES / 16) * (32 / 16);            // 6250
        k_gemm_dual_wmma<<<(waves * 32 + B - 1) / B, B, 0, stream>>>(
            agg2, Wrel2, out1, Wroot2, brel2, out2, N_NODES, 128, 32);
    }
    k_bn_stats<<<128, B, 0, stream>>>(out2, st2, N_NODES, 32);
    k_bn_relu<<<(N_NODES * 32 + B - 1) / B, B, 0, stream>>>(
        out2, st2, gamma2, beta2, N_NODES * 32, 31, 32, 1.0f / N_NODES);

    // pooling + head
    k_pool<<<(N_NODES * 32 + B - 1) / B, B, 0, stream>>>(out2, batch, pool, cnts);
    k_head<<<(N_GRAPHS + B - 1) / B, B, 0, stream>>>(pool, cnts, W1, b1, W2, b2, out);
}